// DRML_27101243637935
// MI455X (gfx1250) — compile-verified
//
#include <hip/hip_runtime.h>
#include <hip/hip_bf16.h>

typedef __attribute__((ext_vector_type(2))) float v2f;
typedef __attribute__((ext_vector_type(8))) float v8f;
typedef unsigned int u32x4 __attribute__((ext_vector_type(4)));
typedef int i32x4 __attribute__((ext_vector_type(4)));
typedef int i32x8 __attribute__((ext_vector_type(8)));

#define EPSBN 1e-5f

#if defined(__HIP_DEVICE_COMPILE__) && __has_builtin(__builtin_amdgcn_tensor_load_to_lds) && __has_builtin(__builtin_amdgcn_s_wait_tensorcnt)
#define HAVE_TDM 1
#else
#define HAVE_TDM 0
#endif

#if HAVE_TDM
// ---------------------------------------------------------------------------
// Stage a contiguous (rows x K) f32 weight panel from global into LDS offset 0
// via the Tensor Data Mover (one wave issues; EXEC is ignored by TDM).
// D# per CDNA5 ISA ch.8: group0 = {flags, lds_addr, global_addr, type=2},
// group1 = {data_size=4B, tensor_dim0=K, tensor_dim1=rows, tile_dim0=K,
//           tile_dim1=rows, tensor_dim0_stride=K}. 2D => groups 2/3 zero.
// ---------------------------------------------------------------------------
__device__ __forceinline__ void tdm_load_weights(const float* w, int K, int rows)
{
    unsigned long long ga = (unsigned long long)(uintptr_t)w;
    u32x4 g0;
    g0.x = 1u;                                           // count=1, user mode
    g0.y = 0u;                                           // lds_addr = dynamic LDS base
    g0.z = (unsigned)(ga & 0xffffffffull);               // global_addr[31:0]
    g0.w = (unsigned)((ga >> 32) & 0x01ffffffull) | (2u << 30); // addr[56:32] | type=2
    i32x8 g1;
    g1[0] = (2 << 16);                                   // workgroup_mask=0, data_size=4B
    g1[1] = (K & 0xffff) << 16;                          // tensor_dim0[15:0]
    g1[2] = ((K >> 16) & 0xffff) | ((rows & 0xffff) << 16); // td0[31:16] | td1[15:0]
    g1[3] = (K & 0xffff) << 16;                          // td1[31:16]=0 | tile_dim0=K
    g1[4] = (rows & 0xffff);                             // tile_dim1=rows, tile_dim2=0
    g1[5] = K;                                           // tensor_dim0_stride[31:0]
    g1[6] = 0;                                           // stride0[47:32] | stride1[15:0]
    g1[7] = 0;                                           // stride1[47:16]
    i32x4 z4 = {0, 0, 0, 0};
#if __clang_major__ >= 23
    i32x8 z8 = {0, 0, 0, 0, 0, 0, 0, 0};
    __builtin_amdgcn_tensor_load_to_lds(g0, g1, z4, z4, z8, 0);
#else
    __builtin_amdgcn_tensor_load_to_lds(g0, g1, z4, z4, 0);
#endif
    __builtin_amdgcn_s_wait_tensorcnt(0);
}
#endif

// ---------------------------------------------------------------------------
// Implicit-GEMM convolution on the fp32 matrix pipe (V_WMMA_F32_16X16X4_F32).
//   GEMM view: M = B*HOUT*WOUT pixels, N = OC (16 or 32), K = CIN*KSZ*KSZ.
//   Per-k gather offset is pixel-independent (VALID conv) -> LDS offset table.
//   Weights (OC,K) staged in LDS (already B^T layout); TDM when K%4==0.
//   Block = 256 thr = 8 waves = (8/NT) pixel-groups x NT n-tiles, NT=OC/16.
// ---------------------------------------------------------------------------
template<bool RELU>
__global__ __launch_bounds__(256) void conv_wmma(
    const float* __restrict__ in, const float* __restrict__ w,
    const float* __restrict__ bias, float* __restrict__ out,
    int M, int OC, int CIN, int KSZ, int STRIDE,
    int HIN, int WIN, int HOUT, int WOUT, int K, int Kp)
{
    extern __shared__ float smem[];
    float* Wt  = smem;                       // [OC * Kp]
    int*  offs = (int*)&smem[OC * Kp];       // [Kp]

    const int tid  = threadIdx.x;
    const int lane = tid & 31;
    const int wave = tid >> 5;
    const int NT   = OC >> 4;                // 1 or 2 n-tiles
    const int pg   = (NT == 2) ? (wave >> 1) : wave;
    const int nt   = (NT == 2) ? (wave & 1) : 0;
    const int half = lane >> 4;
    const int l16  = lane & 15;
    const int m_base = (blockIdx.x * (8 / NT) + pg) * 16;

    // ---- stage weights into LDS ----
    bool staged = false;
#if HAVE_TDM
    if (K == Kp) {
        if (wave == 0) tdm_load_weights(w, K, OC);
        staged = true;
    }
#endif
    if (!staged) {
        for (int e = tid; e < OC * Kp; e += 256) {
            int n = e / Kp, k = e - n * Kp;
            Wt[e] = (k < K) ? w[n * K + k] : 0.0f;
        }
    }
    // ---- im2col offset table (uniform over pixels) ----
    const int KK = KSZ * KSZ;
    for (int k = tid; k < Kp; k += 256) {
        int ic = k / KK, r = k - ic * KK;
        int kh = r / KSZ, kw = r - kh * KSZ;
        offs[k] = (k < K) ? (ic * HIN * WIN + kh * WIN + kw) : 0;
    }
    __syncthreads();

    // ---- per-lane A base pointer (clamped; OOB tiles duplicate last pixel) ----
    const int hw = HOUT * WOUT;
    int mm = m_base + l16; if (mm >= M) mm = M - 1;
    int b  = mm / hw, r2 = mm - b * hw;
    int oh = r2 / WOUT, ow = r2 - oh * WOUT;
    const float* ap = in + (((size_t)b * CIN) * HIN + (size_t)oh * STRIDE) * WIN + (size_t)ow * STRIDE;
    const float* bp = &Wt[(nt * 16 + l16) * Kp];

    v8f acc = {0.f, 0.f, 0.f, 0.f, 0.f, 0.f, 0.f, 0.f};
    for (int k0 = 0; k0 < Kp; k0 += 4) {
        int ks = k0 + half * 2;
        int2 of = *(const int2*)&offs[ks];      // ds_load_b64
        v2f a; a.x = ap[of.x]; a.y = ap[of.y];  // coalesced gathers
        v2f bb = *(const v2f*)&bp[ks];          // ds_load_b64
        acc = __builtin_amdgcn_wmma_f32_16x16x4_f32(
            false, a, false, bb, (short)0, acc, false, false);
    }

    // ---- epilogue: bias (+ReLU), scatter to NCHW ----
    const int n = nt * 16 + l16;
    const float bv = bias[n];
    #pragma unroll
    for (int i = 0; i < 8; ++i) {
        int m = m_base + i + half * 8;
        if (m < M) {
            int b2 = m / hw, r3 = m - b2 * hw;
            int oh2 = r3 / WOUT, ow2 = r3 - oh2 * WOUT;
            float val = acc[i] + bv;
            if (RELU) val = fmaxf(val, 0.0f);
            out[(((size_t)b2 * OC + n) * HOUT + oh2) * WOUT + ow2] = val;
        }
    }
}

// ---------------------------------------------------------------------------
// Region layer stage 1: per-tile eval BN + ReLU, materialized once.
// ---------------------------------------------------------------------------
__global__ __launch_bounds__(256) void region_bnrelu(
    const float* __restrict__ x,
    const float* __restrict__ g, const float* __restrict__ bt,
    const float* __restrict__ m, const float* __restrict__ v,
    float* __restrict__ y, int total)
{
    int idx = blockIdx.x * blockDim.x + threadIdx.x;
    if (idx >= total) return;
    int w = idx % 132; int t = idx / 132;
    int h = t % 132;   t /= 132;
    int c = t % 32;
    int i = h / 18, j = w / 18;
    int p = (i * 8 + j) * 32 + c;
    float val = (x[idx] - m[p]) * (g[p] * rsqrtf(v[p] + EPSBN)) + bt[p];
    y[idx] = fmaxf(val, 0.0f);
}

// ---------------------------------------------------------------------------
// Region layer conv on WMMA: per-tile 3x3 'same' conv (zero pad at tile
// edges) + per-tile bias + residual. blockIdx.y = tile (8x8 grid).
// K = 32*9 = 288. Tile-edge masking via per-k (offset, dih/diw) meta table,
// branch-free v_cndmask (EXEC stays all-ones for WMMA).
// ---------------------------------------------------------------------------
__global__ __launch_bounds__(256) void region_wmma(
    const float* __restrict__ y, const float* __restrict__ x,
    const float* __restrict__ w, const float* __restrict__ cb,
    float* __restrict__ out)
{
    const int K = 288;
    extern __shared__ float smem[];
    float* Wt  = smem;                   // [32 * 288]
    int2* meta = (int2*)&smem[32 * K];   // [288] {offset, (dih+1)|((diw+1)<<16)}

    const int tid  = threadIdx.x;
    const int lane = tid & 31;
    const int wave = tid >> 5;
    const int pg   = wave >> 1;          // 4 pixel groups
    const int nt   = wave & 1;           // 2 n tiles (OC=32)
    const int half = lane >> 4;
    const int l16  = lane & 15;

    const int tile = blockIdx.y;
    const int ti = tile >> 3, tj = tile & 7;
    const int h0 = ti * 18, th = (ti < 7) ? 18 : 6;
    const int w0 = tj * 18, tw = (tj < 7) ? 18 : 6;
    const int Mt = 64 * th * tw;
    const int m_base = (blockIdx.x * 4 + pg) * 16;

    const float* wt_g = w + (size_t)tile * 32 * K;
#if HAVE_TDM
    if (wave == 0) tdm_load_weights(wt_g, K, 32);
#else
    for (int e = tid; e < 32 * K; e += 256) Wt[e] = wt_g[e];
#endif
    for (int k = tid; k < K; k += 256) {
        int ic = k / 9, r = k - ic * 9;
        int kh = r / 3, kw = r - kh * 3;
        int dih = kh - 1, diw = kw - 1;
        int2 e; e.x = ic * 17424 + dih * 132 + diw;
        e.y = (dih + 1) | ((diw + 1) << 16);
        meta[k] = e;
    }
    __syncthreads();

    const int thw = th * tw;
    int mm = m_base + l16; if (mm >= Mt) mm = Mt - 1;
    int b  = mm / thw, r2 = mm - b * thw;
    int lh = r2 / tw, lw = r2 - lh * tw;
    const float* yp = y + ((size_t)(b * 32) * 132 + (h0 + lh)) * 132 + (w0 + lw);
    const float* bp = &Wt[(nt * 16 + l16) * K];

    v8f acc = {0.f, 0.f, 0.f, 0.f, 0.f, 0.f, 0.f, 0.f};
    for (int k0 = 0; k0 < K; k0 += 4) {
        int ks = k0 + half * 2;
        int4 mp = *(const int4*)&meta[ks];  // two meta entries, ds_load_b128
        int dih0 = (mp.y & 0xffff) - 1, diw0 = (mp.y >> 16) - 1;
        int dih1 = (mp.w & 0xffff) - 1, diw1 = (mp.w >> 16) - 1;
        bool v0 = ((unsigned)(lh + dih0) < (unsigned)th) & ((unsigned)(lw + diw0) < (unsigned)tw);
        bool v1 = ((unsigned)(lh + dih1) < (unsigned)th) & ((unsigned)(lw + diw1) < (unsigned)tw);
        v2f a;
        a.x = yp[v0 ? mp.x : 0] * (v0 ? 1.0f : 0.0f);
        a.y = yp[v1 ? mp.z : 0] * (v1 ? 1.0f : 0.0f);
        v2f bb = *(const v2f*)&bp[ks];
        acc = __builtin_amdgcn_wmma_f32_16x16x4_f32(
            false, a, false, bb, (short)0, acc, false, false);
    }

    const int n = nt * 16 + l16;
    const float cbv = cb[tile * 32 + n];
    #pragma unroll
    for (int i = 0; i < 8; ++i) {
        int m = m_base + i + half * 8;
        if (m < Mt) {
            int b2 = m / thw, r3 = m - b2 * thw;
            int lh2 = r3 / tw, lw2 = r3 - lh2 * tw;
            size_t gidx = ((size_t)(b2 * 32 + n) * 132 + (h0 + lh2)) * 132 + (w0 + lw2);
            out[gidx] = acc[i] + cbv + x[gidx];
        }
    }
}

// ---------------------------------------------------------------------------
// Fused ReLU + 2x2 maxpool + BN2.
// ---------------------------------------------------------------------------
__global__ __launch_bounds__(256) void pool_bn(
    const float* __restrict__ in,
    const float* __restrict__ g, const float* __restrict__ bt,
    const float* __restrict__ m, const float* __restrict__ v,
    float* __restrict__ out, int total)
{
    int idx = blockIdx.x * blockDim.x + threadIdx.x;
    if (idx >= total) return;
    int ow = idx % 66; int t = idx / 66;
    int oh = t % 66;   t /= 66;
    int c = t % 32;    int b = t / 32;
    const float* p = in + (((size_t)(b * 32 + c)) * 132 + oh * 2) * 132 + ow * 2;
    float mx = fmaxf(fmaxf(p[0], p[1]), fmaxf(p[132], p[133]));
    mx = fmaxf(mx, 0.0f);
    out[idx] = (mx - m[c]) * (g[c] * rsqrtf(v[c] + EPSBN)) + bt[c];
}

// ---------------------------------------------------------------------------
// FC layer: out[64,N] = X[64,K] * W[N,K]^T + bias  (V_WMMA_F32_16X16X4_F32)
// ---------------------------------------------------------------------------
__global__ __launch_bounds__(256) void fc_wmma(
    const float* __restrict__ X, const float* __restrict__ W,
    const float* __restrict__ bias, float* __restrict__ out,
    int N, int K, int relu)
{
    __shared__ float As[64 * 34];
    __shared__ float Bt[32 * 34];

    const int tid  = threadIdx.x;
    const int lane = tid & 31;
    const int wave = tid >> 5;
    const int m0   = (wave & 3) * 16;
    const int nt   = wave >> 2;
    const int nbase = blockIdx.x * 32;
    const int n0   = nbase + nt * 16;
    const int half = lane >> 4;
    const int l16  = lane & 15;

    v8f acc = {0.f, 0.f, 0.f, 0.f, 0.f, 0.f, 0.f, 0.f};

    for (int kb = 0; kb < K; kb += 32) {
        #pragma unroll
        for (int e = tid; e < 64 * 32; e += 256) {
            int r = e >> 5, k = e & 31;
            As[r * 34 + k] = X[r * K + kb + k];
        }
        #pragma unroll
        for (int e = tid; e < 32 * 32; e += 256) {
            int n = e >> 5, k = e & 31;
            int gn = nbase + n;
            Bt[n * 34 + k] = (gn < N) ? W[(size_t)gn * K + kb + k] : 0.0f;
        }
        if (kb + 32 < K) {
            int pr = nbase + lane;
            if (pr < N) __builtin_prefetch(&W[(size_t)pr * K + kb + 32], 0, 1);
        }
        __syncthreads();

        #pragma unroll
        for (int kk = 0; kk < 32; kk += 4) {
            int ks = kk + half * 2;
            v2f a = *(const v2f*)&As[(m0 + l16) * 34 + ks];
            v2f b = *(const v2f*)&Bt[(nt * 16 + l16) * 34 + ks];
            acc = __builtin_amdgcn_wmma_f32_16x16x4_f32(
                false, a, false, b, (short)0, acc, false, false);
        }
        __syncthreads();
    }

    int n = n0 + l16;
    if (n < N) {
        float bv = bias[n];
        #pragma unroll
        for (int i = 0; i < 8; ++i) {
            int row = m0 + i + half * 8;
            float val = acc[i] + bv;
            if (relu) val = fmaxf(val, 0.0f);
            out[row * N + n] = val;
        }
    }
}

// ---------------------------------------------------------------------------
// Host-side orchestration
// ---------------------------------------------------------------------------
static inline int cdiv(int a, int b) { return (a + b - 1) / b; }

extern "C" void kernel_launch(void* const* d_in, const int* in_sizes, int n_in,
                              void* d_out, int out_size, void* d_ws, size_t ws_size,
                              hipStream_t stream) {
    const float* x       = (const float*)d_in[0];
    const float* conv1_w = (const float*)d_in[1];
    const float* conv1_b = (const float*)d_in[2];
    const float* rg_g    = (const float*)d_in[3];
    const float* rg_b    = (const float*)d_in[4];
    const float* rg_m    = (const float*)d_in[5];
    const float* rg_v    = (const float*)d_in[6];
    const float* rg_w    = (const float*)d_in[7];
    const float* rg_cb   = (const float*)d_in[8];
    const float* bn2_g   = (const float*)d_in[9];
    const float* bn2_b   = (const float*)d_in[10];
    const float* bn2_m   = (const float*)d_in[11];
    const float* bn2_v   = (const float*)d_in[12];
    const float* conv2_w = (const float*)d_in[13];
    const float* conv2_b = (const float*)d_in[14];
    const float* conv3_w = (const float*)d_in[15];
    const float* conv3_b = (const float*)d_in[16];
    const float* conv4_w = (const float*)d_in[17];
    const float* conv4_b = (const float*)d_in[18];
    const float* conv5_w = (const float*)d_in[19];
    const float* conv5_b = (const float*)d_in[20];
    const float* fc1_w   = (const float*)d_in[21];
    const float* fc1_b   = (const float*)d_in[22];
    const float* fc2_w   = (const float*)d_in[23];
    const float* fc2_b   = (const float*)d_in[24];
    const float* fc3_w   = (const float*)d_in[25];
    const float* fc3_b   = (const float*)d_in[26];

    const int B = 64;
    char* ws = (char*)d_ws;
    float* bufA = (float*)(ws);
    float* bufB = (float*)(ws + (size_t)160 * 1024 * 1024);
    float* bufC = (float*)(ws + (size_t)320 * 1024 * 1024);

    // conv1: (64,3,142,142) -> (64,32,132,132)   [WMMA, K=363 (Kp=364)]
    {
        int M = B * 132 * 132;
        int shmem = (32 * 364 + 364) * 4;
        conv_wmma<false><<<cdiv(M, 64), 256, shmem, stream>>>(
            x, conv1_w, conv1_b, bufA, M, 32, 3, 11, 1, 142, 142, 132, 132, 363, 364);
    }
    // region stage 1: BN+ReLU  bufA -> bufB
    {
        int total = B * 32 * 132 * 132;
        region_bnrelu<<<cdiv(total, 256), 256, 0, stream>>>(
            bufA, rg_g, rg_b, rg_m, rg_v, bufB, total);
    }
    // region conv + residual  (bufB, bufA) -> bufC   [WMMA + TDM, K=288]
    {
        int shmem = (32 * 288) * 4 + 288 * 8;
        dim3 grid(cdiv(B * 18 * 18, 64), 64);
        region_wmma<<<grid, 256, shmem, stream>>>(bufB, bufA, rg_w, rg_cb, bufC);
    }
    // relu + maxpool + bn2: bufC -> bufA
    {
        int total = B * 32 * 66 * 66;
        pool_bn<<<cdiv(total, 256), 256, 0, stream>>>(
            bufC, bn2_g, bn2_b, bn2_m, bn2_v, bufA, total);
    }
    // conv2: (64,32,66,66) -> (64,16,59,59)   [WMMA + TDM, K=2048]
    {
        int M = B * 59 * 59;
        int shmem = (16 * 2048 + 2048) * 4;
        conv_wmma<true><<<cdiv(M, 128), 256, shmem, stream>>>(
            bufA, conv2_w, conv2_b, bufB, M, 16, 32, 8, 1, 66, 66, 59, 59, 2048, 2048);
    }
    // conv3: (64,16,59,59) -> (64,16,52,52)   [WMMA + TDM, K=1024]
    {
        int M = B * 52 * 52;
        int shmem = (16 * 1024 + 1024) * 4;
        conv_wmma<true><<<cdiv(M, 128), 256, shmem, stream>>>(
            bufB, conv3_w, conv3_b, bufA, M, 16, 16, 8, 1, 59, 59, 52, 52, 1024, 1024);
    }
    // conv4: stride 2 -> (64,16,24,24)        [WMMA + TDM, K=576]
    {
        int M = B * 24 * 24;
        int shmem = (16 * 576 + 576) * 4;
        conv_wmma<true><<<cdiv(M, 128), 256, shmem, stream>>>(
            bufA, conv4_w, conv4_b, bufB, M, 16, 16, 6, 2, 52, 52, 24, 24, 576, 576);
    }
    // conv5: -> (64,16,20,20) == flatten (64,6400)   [WMMA + TDM, K=400]
    {
        int M = B * 20 * 20;
        int shmem = (16 * 400 + 400) * 4;
        conv_wmma<true><<<cdiv(M, 128), 256, shmem, stream>>>(
            bufB, conv5_w, conv5_b, bufA, M, 16, 16, 5, 1, 24, 24, 20, 20, 400, 400);
    }
    // fc1..fc3 [WMMA]
    fc_wmma<<<cdiv(4096, 32), 256, 0, stream>>>(bufA, fc1_w, fc1_b, bufB, 4096, 6400, 1);
    fc_wmma<<<cdiv(2048, 32), 256, 0, stream>>>(bufB, fc2_w, fc2_b, bufA, 2048, 4096, 1);
    fc_wmma<<<1, 256, 0, stream>>>(bufA, fc3_w, fc3_b, (float*)d_out, 12, 2048, 0);
}